// PauliGate_36000415875236
// MI455X (gfx1250) — compile-verified
//
#include <hip/hip_runtime.h>
#include <cstdint>

// ---------------------------------------------------------------------------
// Pauli gate on qubit 12 of a 25-qubit (batch=4) real f32 state vector.
// out[a, site] = sum_k P[a][k] * state[k, site]   (general 2x2 real gate)
//
// Pure streaming permutation: 1 GiB of traffic, ~46 us at 23.3 TB/s HBM.
// Data path: 128-bit coalesced accesses, non-temporal stores (stream does
// not fit 192 MB L2, each element touched once), and CDNA5 async
// global->LDS loads (ASYNCcnt) for the input stream.
// ---------------------------------------------------------------------------

typedef float v4f __attribute__((ext_vector_type(4)));
typedef int   v4i __attribute__((vector_size(4 * sizeof(int))));  // matches builtin param

#define AS1 __attribute__((address_space(1)))
#define AS3 __attribute__((address_space(3)))

#if __has_builtin(__builtin_amdgcn_global_load_async_to_lds_b128)
#define USE_ASYNC_LDS 1
#else
#define USE_ASYNC_LDS 0
#endif

// Reference constants: N_QUBITS=25, TARGET_QUBIT=12, BATCH=4.
// Element stride of the target-qubit axis = 2^(25-1-12) * 4 = 16384 elements
// = 4096 float4s => bit 12 in float4-index space.
constexpr unsigned kLowBits   = 12;
constexpr unsigned kQStride4  = 1u << kLowBits;   // float4 units
constexpr int      kBlock     = 256;              // 8 waves (wave32)

__global__ __launch_bounds__(kBlock) void pauli_gate_swap(
    const float* __restrict__ state,
    const float* __restrict__ pauli,
    float* __restrict__ out,
    unsigned npair4)   // number of float4 pairs = N/8 = 2^24
{
    // 2x2 gate coefficients: uniform -> scalar loads.
    const float p00 = pauli[0];
    const float p01 = pauli[1];
    const float p10 = pauli[2];
    const float p11 = pauli[3];

    const v4f* __restrict__ in4  = (const v4f*)state;
    v4f* __restrict__       out4 = (v4f*)out;

    unsigned tid     = blockIdx.x * kBlock + threadIdx.x;
    unsigned gstride = gridDim.x * kBlock;

#if USE_ASYNC_LDS
    // Per-lane private staging slots: 2 x 16B per thread = 8 KB per block
    // (LDS pool is 320 KB per WGP; negligible occupancy impact).
    __shared__ v4f lds_stage[kBlock * 2];
    v4f* myslot = &lds_stage[threadIdx.x * 2u];
    // LDS-address operand must carry the shared address space: a real
    // addrspacecast (generic flat -> LDS offset) is required here.
    AS3 v4i* l0 = (AS3 v4i*)(myslot + 0);
    AS3 v4i* l1 = (AS3 v4i*)(myslot + 1);
#endif

    for (unsigned t = tid; t < npair4; t += gstride) {
        // Split flat pair index around the qubit bit.
        unsigned low  = ((t >> kLowBits) << (kLowBits + 1)) | (t & (kQStride4 - 1u));
        unsigned high = low + kQStride4;

        v4f s0, s1;
#if USE_ASYNC_LDS
        // WAR fence for slot reuse: async LDS writes are NOT ordered with the
        // wave's DS pipe, so make sure the previous iteration's ds_loads have
        // drained before overwriting the slots.  Free in steady state.
        asm volatile("s_wait_dscnt 0" ::: "memory");

        // Global pointers: generic and AS1 share the raw 64-bit address on
        // AMDGPU, so inttoptr gives a correctly-typed, correctly-valued arg
        // (also drops const, which the builtin parameter requires).
        AS1 v4i* g0 = (AS1 v4i*)(uintptr_t)(in4 + low);
        AS1 v4i* g1 = (AS1 v4i*)(uintptr_t)(in4 + high);

        // CDNA5 async streaming path: global -> LDS without VGPR staging,
        // tracked by ASYNCcnt.  Each lane writes its own private LDS slot,
        // so a per-wave s_wait_asynccnt suffices (no barrier needed).
        __builtin_amdgcn_global_load_async_to_lds_b128(g0, l0, 0, 0);
        __builtin_amdgcn_global_load_async_to_lds_b128(g1, l1, 0, 0);
        // Hard ordering point: compiler may not model the async LDS write,
        // so fence with inline asm + memory clobber.
        asm volatile("s_wait_asynccnt 0" ::: "memory");
        s0 = myslot[0];
        s1 = myslot[1];
#else
        // Fallback: direct 128-bit non-temporal loads (th:NT).
        s0 = __builtin_nontemporal_load(in4 + low);
        s1 = __builtin_nontemporal_load(in4 + high);
#endif

        // General 2x2 gate (for Pauli X this is the half-space swap).
        v4f o0 = p00 * s0 + p01 * s1;
        v4f o1 = p10 * s0 + p11 * s1;

        // Streaming stores: never re-read, bypass temporal caching.
        __builtin_nontemporal_store(o0, out4 + low);
        __builtin_nontemporal_store(o1, out4 + high);
    }
}

extern "C" void kernel_launch(void* const* d_in, const int* in_sizes, int n_in,
                              void* d_out, int out_size, void* d_ws, size_t ws_size,
                              hipStream_t stream) {
    (void)n_in; (void)out_size; (void)d_ws; (void)ws_size;

    const float* state = (const float*)d_in[0];   // 2^27 f32
    const float* pauli = (const float*)d_in[1];   // 2x2 f32 row-major
    float* out         = (float*)d_out;           // 2^27 f32

    long long n      = (long long)in_sizes[0];    // 134217728
    unsigned  npair4 = (unsigned)(n / 8);         // 2^24 float4 pairs

    // One pair (32 B in + 32 B out) per thread, exact cover.
    unsigned blocks = (npair4 + kBlock - 1) / kBlock;   // 65536
    pauli_gate_swap<<<dim3(blocks), dim3(kBlock), 0, stream>>>(
        state, pauli, out, npair4);
}